// Resort_16209206575145
// MI455X (gfx1250) — compile-verified
//
#include <hip/hip_runtime.h>
#include <hip/hip_bf16.h>

// Problem constants (from reference: img = (32, 2048, 690) f32)
constexpr int kB    = 32;
constexpr int kC    = 2048;
constexpr int kW    = 690;
constexpr int kNBC  = kB * kC;          // 65536 rows
constexpr int kChunks = 128;            // row chunks for deterministic reduction
constexpr int kRowsPerChunk = kNBC / kChunks;  // 512
constexpr int kTiles = (kW + 15) / 16;  // 44 column tiles of 16
constexpr int kSeg  = 2 * kW + 1;       // 1381 segments (lengths array)
constexpr int kHalf = (kSeg + 1) / 2;   // 691 threefry cipher instances

typedef float v2f __attribute__((ext_vector_type(2)));
typedef float v8f __attribute__((ext_vector_type(8)));

#if __has_builtin(__builtin_amdgcn_wmma_f32_16x16x4_f32)
#define HAVE_WMMA_F32X4 1
#else
#define HAVE_WMMA_F32X4 0
#endif

#if __has_builtin(__builtin_amdgcn_global_load_async_to_lds_b32)
#define HAVE_ASYNC_LDS 1
typedef __attribute__((address_space(3))) int lds_i32;
typedef __attribute__((address_space(1))) int gbl_i32;
#else
#define HAVE_ASYNC_LDS 0
#endif

// ---------------------------------------------------------------------------
// Kernel A: column partial sums via ones-matrix WMMA:  D = 1(16x4) * B(4x16) + C
// Every lane of D holds the column sum of the 4 rows fed in B. Out-of-range
// lanes (last tile only) load a clamped in-bounds address unconditionally:
// with A == ones, garbage there only lands in D columns we never write, so the
// inner loop is branch-free (2 global_load_b32 + 1 v_wmma per 4 rows).
// Grid: (11, 128), block 128 (4 waves); wave w of block handles tile 4*bx+w.
// ---------------------------------------------------------------------------
__global__ void __launch_bounds__(128)
colsum_wmma_kernel(const float* __restrict__ img, float* __restrict__ partial) {
    const int wave = threadIdx.x >> 5;
    const int lane = threadIdx.x & 31;
    const int tile = blockIdx.x * 4 + wave;        // 0..43
    const int n0   = tile * 16;
    const int n    = n0 + (lane & 15);
    const bool valid = (n < kW);
    const int ncl  = valid ? n : 0;                // clamped, always in-bounds
    const int rsel = (lane >> 4) * 2;              // lanes 0-15 -> rows +0/+1, 16-31 -> +2/+3
    const int row0 = blockIdx.y * kRowsPerChunk;

    const float* base = img + (size_t)(row0 + rsel) * kW + ncl;

#if HAVE_WMMA_F32X4
    v8f acc = {0.f, 0.f, 0.f, 0.f, 0.f, 0.f, 0.f, 0.f};
    v2f a = {1.0f, 1.0f};                          // A = ones(16x4)
#else
    float accs = 0.0f;
#endif

    for (int r = 0; r < kRowsPerChunk; r += 4) {
        const float* p = base + (size_t)r * kW;
        float b0 = p[0];                           // row r + rsel
        float b1 = p[kW];                          // row r + rsel + 1
#if HAVE_WMMA_F32X4
        v2f b = {b0, b1};
        acc = __builtin_amdgcn_wmma_f32_16x16x4_f32(
            /*neg_a=*/false, a, /*neg_b=*/false, b,
            /*c_mod=*/(short)0, acc, /*reuse_a=*/false, /*reuse_b=*/false);
#else
        accs += b0 + b1;
#endif
    }

#if HAVE_WMMA_F32X4
    float v = acc[0];   // D[.,n] replicated: full 16-col sums of this row chunk
#else
    float other = __shfl(accs, (lane & 15) + 16, 32);  // rows +2/+3 half
    float v = accs + other;
#endif
    if (lane < 16 && valid) {
        partial[(size_t)blockIdx.y * kW + n] = v;
    }
}

// ---------------------------------------------------------------------------
// Threefry-2x32 (exact JAX replication, 20 rounds) and uniform conversion
// ---------------------------------------------------------------------------
__device__ __forceinline__ unsigned rotl32(unsigned x, int d) {
    return (x << d) | (x >> (32 - d));
}
__device__ __forceinline__ void threefry2x32(unsigned k0, unsigned k1,
                                             unsigned& x0, unsigned& x1) {
    const unsigned ks0 = k0, ks1 = k1, ks2 = k0 ^ k1 ^ 0x1BD11BDAu;
    const int rotA[4] = {13, 15, 26, 6};
    const int rotB[4] = {17, 29, 16, 24};
    x0 += ks0; x1 += ks1;
#pragma unroll
    for (int i = 0; i < 4; ++i) { x0 += x1; x1 = rotl32(x1, rotA[i]); x1 ^= x0; }
    x0 += ks1; x1 += ks2 + 1u;
#pragma unroll
    for (int i = 0; i < 4; ++i) { x0 += x1; x1 = rotl32(x1, rotB[i]); x1 ^= x0; }
    x0 += ks2; x1 += ks0 + 2u;
#pragma unroll
    for (int i = 0; i < 4; ++i) { x0 += x1; x1 = rotl32(x1, rotA[i]); x1 ^= x0; }
    x0 += ks0; x1 += ks1 + 3u;
#pragma unroll
    for (int i = 0; i < 4; ++i) { x0 += x1; x1 = rotl32(x1, rotB[i]); x1 ^= x0; }
    x0 += ks1; x1 += ks2 + 4u;
#pragma unroll
    for (int i = 0; i < 4; ++i) { x0 += x1; x1 = rotl32(x1, rotA[i]); x1 ^= x0; }
    x0 += ks2; x1 += ks0 + 5u;
}
__device__ __forceinline__ float bits_to_uniform(unsigned b) {
    unsigned fb = (b >> 9) | 0x3F800000u;   // mantissa fill, [1,2)
    float f;
    __builtin_memcpy(&f, &fb, 4);
    return f - 1.0f;                         // [0,1), bit-exact vs JAX
}

// ---------------------------------------------------------------------------
// Kernel B: all scalar/segment logic in one block (deterministic).
// ---------------------------------------------------------------------------
__global__ void __launch_bounds__(1024)
segment_logic_kernel(const float* __restrict__ partial, int* __restrict__ src_out) {
    __shared__ float col[kW];
    __shared__ float cs[kW + 1];
    __shared__ int   m[kW];
    __shared__ int   cuts[2 * kW + 2];     // 1382
    __shared__ int   hist[kW + 1];         // counting sort bins 0..W
    __shared__ int   starts[kSeg];
    __shared__ int   lens[kSeg];
    __shared__ float u[kSeg];
    __shared__ int   perm[kSeg];
    __shared__ int   ps[kSeg];
    __shared__ int   pl[kSeg];
    __shared__ int   cum[kSeg];

    const int tid = threadIdx.x;

    // 1) fixed-order reduction of partial column sums
    for (int w = tid; w < kW; w += blockDim.x) {
        float s = 0.0f;
        for (int k = 0; k < kChunks; ++k) s += partial[(size_t)k * kW + w];
        col[w] = s;
    }
    for (int v = tid; v <= kW; v += blockDim.x) hist[v] = 0;
    __syncthreads();

    // 2) sequential cumsum (matches jnp.cumsum order)
    if (tid == 0) {
        float s = 0.0f;
        cs[0] = 0.0f;
        for (int w = 0; w < kW; ++w) { s += col[w]; cs[w + 1] = s; }
    }
    __syncthreads();

    // 3) windowed mean mask
    const float gmean = cs[kW] / ((float)kNBC * (float)kW);
    for (int i = tid; i < kW; i += blockDim.x) {
        int end = (i + 4 < kW) ? i + 4 : kW;
        float wsum  = cs[end] - cs[i];
        float wmean = wsum / ((float)(end - i) * (float)kNBC);
        m[i] = (wmean >= gmean) ? 1 : 0;
    }
    __syncthreads();

    // 4) cuts = [0, W] ++ where(is_start, idx, W) ++ where(is_last, idx, W)
    for (int i = tid; i < kW; i += blockDim.x) {
        int prev = (i == 0)      ? 0 : m[i - 1];
        int next = (i == kW - 1) ? 0 : m[i + 1];
        cuts[2 + i]      = (m[i] && !prev) ? i : kW;
        cuts[2 + kW + i] = (m[i] && !next) ? i : kW;
    }
    if (tid == 0) { cuts[0] = 0; cuts[1] = kW; }
    __syncthreads();

    // 5) sort cuts: counting sort over values 0..W (serial, tiny)
    if (tid == 0) {
        for (int i = 0; i < 2 * kW + 2; ++i) hist[cuts[i]]++;
        int pos = 0;
        for (int v = 0; v <= kW; ++v)
            for (int k = 0; k < hist[v]; ++k) cuts[pos++] = v;
    }
    __syncthreads();

    // 6) starts / lengths
    for (int i = tid; i < kSeg; i += blockDim.x) {
        starts[i] = cuts[i];
        lens[i]   = cuts[i + 1] - cuts[i];
    }

    // 7) JAX uniform(key(42), (1381,)) via threefry2x32; odd-size pad like JAX
    for (int j = tid; j < kHalf; j += blockDim.x) {
        unsigned x0 = (unsigned)j;
        unsigned x1 = (j == kHalf - 1) ? 0u : (unsigned)(kHalf + j);
        threefry2x32(0u, 42u, x0, x1);
        u[j] = bits_to_uniform(x0);
        if (kHalf + j < kSeg) u[kHalf + j] = bits_to_uniform(x1);
    }
    __syncthreads();

    // 8) stable argsort via rank (bit-exact perm)
    for (int i = tid; i < kSeg; i += blockDim.x) {
        float ui = u[i];
        int r = 0;
        for (int j = 0; j < kSeg; ++j) {
            float uj = u[j];
            r += (uj < ui) || (uj == ui && j < i);
        }
        perm[r] = i;
    }
    __syncthreads();

    // 9) permuted starts/lengths
    for (int i = tid; i < kSeg; i += blockDim.x) {
        ps[i] = starts[perm[i]];
        pl[i] = lens[perm[i]];
    }
    __syncthreads();
    if (tid == 0) {
        int s = 0;
        for (int i = 0; i < kSeg; ++i) { s += pl[i]; cum[i] = s; }
    }
    __syncthreads();

    // 10) src[pos]: searchsorted(cum, pos, 'right') then offset into segment
    for (int pos = tid; pos < kW; pos += blockDim.x) {
        int lo = 0, hi = kSeg - 1;
        while (lo < hi) {
            int mid = (lo + hi) >> 1;
            if (cum[mid] > pos) hi = mid; else lo = mid + 1;
        }
        src_out[pos] = ps[lo] + (pos - (cum[lo] - pl[lo]));
    }
}

// ---------------------------------------------------------------------------
// Kernel C: gather along W. img rows should be L2-resident (181 MB < 192 MB L2,
// warmed by kernel A); NT stores keep output from evicting img. src table is
// staged into LDS via CDNA5 async-to-LDS (ASYNCcnt) when available.
// ---------------------------------------------------------------------------
constexpr int kRowsPerGatherBlock = 32;

__global__ void __launch_bounds__(256)
gather_kernel(const float* __restrict__ img, const int* __restrict__ src,
              float* __restrict__ out) {
    __shared__ int s_src[kW];

#if HAVE_ASYNC_LDS
    for (int i = threadIdx.x; i < kW; i += blockDim.x) {
        __builtin_amdgcn_global_load_async_to_lds_b32(
            (gbl_i32*)(src + i), (lds_i32*)(s_src + i), 0, 0);
    }
#if __has_builtin(__builtin_amdgcn_s_wait_asynccnt)
    __builtin_amdgcn_s_wait_asynccnt(0);
#else
    asm volatile("s_wait_asynccnt 0x0" ::: "memory");
#endif
#else
    for (int i = threadIdx.x; i < kW; i += blockDim.x) s_src[i] = src[i];
#endif
    __syncthreads();

    const size_t row0 = (size_t)blockIdx.x * kRowsPerGatherBlock;
    for (int r = 0; r < kRowsPerGatherBlock; ++r) {
        const size_t base = (row0 + r) * (size_t)kW;
        for (int p = threadIdx.x; p < kW; p += blockDim.x) {
            float v = img[base + s_src[p]];
            __builtin_nontemporal_store(v, &out[base + p]);
        }
    }
}

// ---------------------------------------------------------------------------
extern "C" void kernel_launch(void* const* d_in, const int* in_sizes, int n_in,
                              void* d_out, int out_size, void* d_ws, size_t ws_size,
                              hipStream_t stream) {
    (void)in_sizes; (void)n_in; (void)out_size; (void)ws_size;
    const float* img = (const float*)d_in[0];
    float* out = (float*)d_out;

    float* partial = (float*)d_ws;                                    // 128*690 f32
    int*   src     = (int*)((char*)d_ws + (size_t)kChunks * kW * sizeof(float));

    dim3 gridA(kTiles / 4, kChunks);   // (11, 128); 4 waves/block -> 44 tiles
    colsum_wmma_kernel<<<gridA, 128, 0, stream>>>(img, partial);

    segment_logic_kernel<<<1, 1024, 0, stream>>>(partial, src);

    gather_kernel<<<kNBC / kRowsPerGatherBlock, 256, 0, stream>>>(img, src, out);
}